// GNN_Model_59974923321554
// MI455X (gfx1250) — compile-verified
//
#include <hip/hip_runtime.h>
#include <hip/hip_bf16.h>

// ---------------------------------------------------------------------------
// GNN model for MI455X (gfx1250, wave32). All dense math via
// V_WMMA_F32_16X16X4_F32; edge scatter via native f32 global atomics.
// ---------------------------------------------------------------------------

typedef __attribute__((ext_vector_type(2))) float v2f;
typedef __attribute__((ext_vector_type(8))) float v8f;

#define NPN   65536      // B * PNODE_NUM
#define NFN   1024       // B * FNODE_NUM
#define E2N   1048576    // E / 2
#define GD    128
#define LIN0  4096
#define LOUT1 4089
#define LOUT2 4082
#define LOUT3 4075

__device__ __forceinline__ void atomAddF(float* p, float v) {
    __hip_atomic_fetch_add(p, v, __ATOMIC_RELAXED, __HIP_MEMORY_SCOPE_AGENT);
}

// ---------------------------------------------------------------------------
// Generic  out = relu?( A(MxK) @ W(NxK)^T [+ A2(MxK2) @ W2(NxK2)^T] + bias + addend )
// One wave per 16x16 output tile. M,K,K2 assumed multiples of 16/4; N guarded.
// ---------------------------------------------------------------------------
__global__ void __launch_bounds__(256)
gemm_nt_wmma(const float* __restrict__ A, const float* __restrict__ W,
             const float* __restrict__ bias, const float* __restrict__ addend,
             const float* __restrict__ A2, const float* __restrict__ W2,
             float* __restrict__ out,
             int M, int N, int K, int K2, int relu)
{
    const int ntiles = (N + 15) >> 4;
    const int mtiles = (M + 15) >> 4;
    const int wave   = blockIdx.x * (blockDim.x >> 5) + (threadIdx.x >> 5);
    if (wave >= ntiles * mtiles) return;          // wave-uniform: EXEC stays all-1s
    const int mt = wave / ntiles, nt = wave % ntiles;
    const int lane = threadIdx.x & 31;
    const int half = lane >> 4, lm = lane & 15;

    int row = mt * 16 + lm; if (row >= M) row = M - 1;     // clamp (rows independent)
    const int col  = nt * 16 + lm;
    const int colc = (col < N) ? col : (N - 1);            // clamp (cols independent)

    const float* __restrict__ arow = A + (size_t)row  * K;
    const float* __restrict__ wrow = W + (size_t)colc * K;

    v8f c = {};
    for (int k = 0; k < K; k += 4) {
        if ((k & 255) == 0) {                // exercise global_prefetch_b8
            __builtin_prefetch(wrow + k + 256, 0, 0);
            __builtin_prefetch(arow + k + 256, 0, 0);
        }
        v2f a, b;
        a[0] = arow[k + 2 * half];  a[1] = arow[k + 2 * half + 1];
        b[0] = wrow[k + 2 * half];  b[1] = wrow[k + 2 * half + 1];
        c = __builtin_amdgcn_wmma_f32_16x16x4_f32(false, a, false, b, 0, c, false, false);
    }
    if (A2) {
        const float* __restrict__ arow2 = A2 + (size_t)row  * K2;
        const float* __restrict__ wrow2 = W2 + (size_t)colc * K2;
        for (int k = 0; k < K2; k += 4) {
            v2f a, b;
            a[0] = arow2[k + 2 * half];  a[1] = arow2[k + 2 * half + 1];
            b[0] = wrow2[k + 2 * half];  b[1] = wrow2[k + 2 * half + 1];
            c = __builtin_amdgcn_wmma_f32_16x16x4_f32(false, a, false, b, 0, c, false, false);
        }
    }

    const float bv = bias ? bias[colc] : 0.0f;
    #pragma unroll
    for (int v = 0; v < 8; ++v) {
        const int r = mt * 16 + v + 8 * half;   // D layout: VGPR v -> M=v (+8 for hi half)
        if (r < M && col < N) {
            float val = c[v] + bv;
            if (addend) val += addend[(size_t)r * N + col];
            if (relu)   val = fmaxf(val, 0.0f);
            out[(size_t)r * N + col] = val;
        }
    }
}

// ---------------------------------------------------------------------------
// Implicit-GEMM 1D conv (VALID), NCH/OIH. N=64 out channels, K=Cin*8.
// out[b,o,l] = relu( sum_{c,t} in[b,c,l+t] * W[o,c,t] + bias[o] )
// ---------------------------------------------------------------------------
__global__ void __launch_bounds__(256)
conv1d_wmma(const float* __restrict__ in, const float* __restrict__ W,
            const float* __restrict__ bias, float* __restrict__ out,
            int Cin, int Lin, int Lout)
{
    const int bidx   = blockIdx.y;               // batch
    const int ntiles = 4;                        // 64 / 16
    const int mtiles = (Lout + 15) >> 4;
    const int wave   = blockIdx.x * (blockDim.x >> 5) + (threadIdx.x >> 5);
    if (wave >= mtiles * ntiles) return;
    const int mt = wave / ntiles, nt = wave % ntiles;
    const int lane = threadIdx.x & 31;
    const int half = lane >> 4, lm = lane & 15;

    const int pos  = mt * 16 + lm;
    const int posc = (pos < Lout) ? pos : 0;     // clamp; rows independent
    const int col  = nt * 16 + lm;               // out channel
    const int K    = Cin * 8;

    const float* __restrict__ inb  = in + (size_t)bidx * Cin * Lin;
    const float* __restrict__ wrow = W  + (size_t)col * K;

    v8f c = {};
    for (int k = 0; k < K; k += 4) {
        const int kk0 = k + 2 * half, kk1 = kk0 + 1;
        v2f a, b;
        a[0] = inb[(kk0 >> 3) * Lin + posc + (kk0 & 7)];
        a[1] = inb[(kk1 >> 3) * Lin + posc + (kk1 & 7)];
        b[0] = wrow[kk0];
        b[1] = wrow[kk1];
        c = __builtin_amdgcn_wmma_f32_16x16x4_f32(false, a, false, b, 0, c, false, false);
    }

    const float bv = bias[col];
    float* __restrict__ outb = out + ((size_t)bidx * 64 + col) * Lout;
    #pragma unroll
    for (int v = 0; v < 8; ++v) {
        const int r = mt * 16 + v + 8 * half;
        if (r < Lout) outb[r] = fmaxf(c[v] + bv, 0.0f);
    }
}

// ---------------------------------------------------------------------------
// Split-K GEMM with f32-atomic reduction (d1: M=16, N=100, K=260800).
// ---------------------------------------------------------------------------
__global__ void __launch_bounds__(256)
gemm_nt_splitk(const float* __restrict__ A, const float* __restrict__ W,
               float* __restrict__ acc, int N, int K, int CHUNK)
{
    const int ntiles = (N + 15) >> 4;
    const int chunks = (K + CHUNK - 1) / CHUNK;
    const int wave   = blockIdx.x * (blockDim.x >> 5) + (threadIdx.x >> 5);
    if (wave >= ntiles * chunks) return;
    const int nt = wave / chunks, ch = wave % chunks;
    const int lane = threadIdx.x & 31;
    const int half = lane >> 4, lm = lane & 15;

    const int col  = nt * 16 + lm;
    const int colc = (col < N) ? col : (N - 1);
    const int k0 = ch * CHUNK;
    const int k1 = (k0 + CHUNK < K) ? (k0 + CHUNK) : K;

    const float* __restrict__ arow = A + (size_t)lm * K;      // M = 16 rows
    const float* __restrict__ wrow = W + (size_t)colc * K;

    v8f c = {};
    for (int k = k0; k < k1; k += 4) {
        v2f a, b;
        a[0] = arow[k + 2 * half];  a[1] = arow[k + 2 * half + 1];
        b[0] = wrow[k + 2 * half];  b[1] = wrow[k + 2 * half + 1];
        c = __builtin_amdgcn_wmma_f32_16x16x4_f32(false, a, false, b, 0, c, false, false);
    }
    #pragma unroll
    for (int v = 0; v < 8; ++v) {
        const int r = v + 8 * half;                 // row in [0,16)
        if (col < N) atomAddF(&acc[(size_t)r * N + col], c[v]);
    }
}

// --------------------------- elementwise helpers ---------------------------

__global__ void zero_f32(float* __restrict__ p, long long n) {
    long long i = (long long)blockIdx.x * blockDim.x + threadIdx.x;
    if (i < n) p[i] = 0.0f;
}

// dim-3 edge scatter (gather src row, atomically accumulate into dst row + count)
__global__ void scatter_mean3(const float* __restrict__ src,
                              const int* __restrict__ sidx, const int* __restrict__ didx,
                              int off, float* __restrict__ sum, float* __restrict__ cnt)
{
    const int j = blockIdx.x * blockDim.x + threadIdx.x;
    if (j >= E2N) return;
    const int s = sidx[2 * j + off];
    const int d = didx[2 * j + off];
    atomAddF(&sum[(size_t)d * 3 + 0], src[(size_t)s * 3 + 0]);
    atomAddF(&sum[(size_t)d * 3 + 1], src[(size_t)s * 3 + 1]);
    atomAddF(&sum[(size_t)d * 3 + 2], src[(size_t)s * 3 + 2]);
    atomAddF(&cnt[d], 1.0f);
}

// 128-dim edge scatter: 8 threads per edge, 16 floats each
__global__ void scatter_mean128(const float* __restrict__ src,
                                const int* __restrict__ sidx, const int* __restrict__ didx,
                                int off, float* __restrict__ sum, float* __restrict__ cnt)
{
    const long long gid = (long long)blockIdx.x * blockDim.x + threadIdx.x;
    const int j = (int)(gid >> 3);
    if (j >= E2N) return;
    const int part = (int)(gid & 7);
    const int s = sidx[2 * j + off];
    const int d = didx[2 * j + off];
    const float* __restrict__ sp = src + (size_t)s * GD + part * 16;
    float* __restrict__ dp = sum + (size_t)d * GD + part * 16;
    #pragma unroll
    for (int c = 0; c < 16; ++c) atomAddF(&dp[c], sp[c]);
    if (part == 0) atomAddF(&cnt[d], 1.0f);
}

__global__ void mean_div(float* __restrict__ sum, const float* __restrict__ cnt,
                         long long rows, int dim)
{
    long long i = (long long)blockIdx.x * blockDim.x + threadIdx.x;
    if (i >= rows * dim) return;
    sum[i] /= fmaxf(cnt[i / dim], 1.0f);
}

// SAGE combine for K=3 inputs (s1_0): out = relu(mean@Wl^T + bl + xd@Wr^T)
__global__ void sage_combine_small(const float* __restrict__ sum3, const float* __restrict__ cnt,
                                   const float* __restrict__ xd3,
                                   const float* __restrict__ Wl, const float* __restrict__ bl,
                                   const float* __restrict__ Wr,
                                   float* __restrict__ out, long long rows)
{
    long long i = (long long)blockIdx.x * blockDim.x + threadIdx.x;
    if (i >= rows * GD) return;
    const long long m = i >> 7;
    const int n = (int)(i & 127);
    const float ic = 1.0f / fmaxf(cnt[m], 1.0f);
    float acc = bl[n];
    #pragma unroll
    for (int k = 0; k < 3; ++k)
        acc += (sum3[m * 3 + k] * ic) * Wl[n * 3 + k] + xd3[m * 3 + k] * Wr[n * 3 + k];
    out[i] = fmaxf(acc, 0.0f);
}

// addend = A3(rows x 3) @ Wr(128 x 3)^T  (small-K side of s2_0)
__global__ void smallk_nt(const float* __restrict__ A3, const float* __restrict__ Wr,
                          float* __restrict__ out, long long rows)
{
    long long i = (long long)blockIdx.x * blockDim.x + threadIdx.x;
    if (i >= rows * GD) return;
    const long long m = i >> 7;
    const int n = (int)(i & 127);
    float acc = 0.0f;
    #pragma unroll
    for (int k = 0; k < 3; ++k) acc += A3[m * 3 + k] * Wr[n * 3 + k];
    out[i] = acc;
}

// per-row LayerNorm over 128 features, in place (LDS tree reduction)
__global__ void __launch_bounds__(128)
layernorm_rows(float* __restrict__ x, const float* __restrict__ g,
               const float* __restrict__ b)
{
    __shared__ float red[GD];
    float* __restrict__ xr = x + (size_t)blockIdx.x * GD;
    const int t = threadIdx.x;
    const float v = xr[t];
    red[t] = v; __syncthreads();
    for (int s = 64; s > 0; s >>= 1) { if (t < s) red[t] += red[t + s]; __syncthreads(); }
    const float mean = red[0] * (1.0f / GD);
    __syncthreads();
    const float d = v - mean;
    red[t] = d * d; __syncthreads();
    for (int s = 64; s > 0; s >>= 1) { if (t < s) red[t] += red[t + s]; __syncthreads(); }
    const float inv = rsqrtf(red[0] * (1.0f / GD) + 1e-5f);
    xr[t] = d * inv * g[t] + b[t];
}

__global__ void bias_relu(const float* __restrict__ acc, const float* __restrict__ b,
                          float* __restrict__ out, int M, int N)
{
    const int i = blockIdx.x * blockDim.x + threadIdx.x;
    if (i >= M * N) return;
    out[i] = fmaxf(acc[i] + b[i % N], 0.0f);
}

// final 100->2 dense + softmax, one thread per batch row
__global__ void d2_softmax(const float* __restrict__ x, const float* __restrict__ W,
                           const float* __restrict__ b, float* __restrict__ out)
{
    const int m = threadIdx.x;
    if (m >= 16) return;
    float l0 = b[0], l1 = b[1];
    for (int k = 0; k < 100; ++k) {
        const float v = x[m * 100 + k];
        l0 += v * W[k];
        l1 += v * W[100 + k];
    }
    const float mx = fmaxf(l0, l1);
    const float e0 = __expf(l0 - mx), e1 = __expf(l1 - mx);
    const float s = e0 + e1;
    out[m * 2 + 0] = e0 / s;
    out[m * 2 + 1] = e1 / s;
}

// ---------------------------------------------------------------------------
// Orchestration
// ---------------------------------------------------------------------------
static inline dim3 waveGrid(long long waves) { return dim3((unsigned)((waves + 7) / 8)); }
static inline dim3 elemGrid(long long n)     { return dim3((unsigned)((n + 255) / 256)); }

extern "C" void kernel_launch(void* const* d_in, const int* in_sizes, int n_in,
                              void* d_out, int out_size, void* d_ws, size_t ws_size,
                              hipStream_t stream)
{
    (void)in_sizes; (void)n_in; (void)out_size; (void)ws_size;

    const float* x_src = (const float*)d_in[0];        // (1024,) == (16,64)
    const float* x_dst = (const float*)d_in[1];        // (65536,3) == (16,12288)
    const int*   edge  = (const int*)  d_in[2];        // (2, E) int32
    int pi = 3;
    const float* pnode_W = (const float*)d_in[pi++];   // (12288,12288)
    const float* pnode_b = (const float*)d_in[pi++];
    const float* fnode_W = (const float*)d_in[pi++];   // (192,64)
    const float* fnode_b = (const float*)d_in[pi++];
    const float* s10_Wl  = (const float*)d_in[pi++];   // (128,3)
    const float* s10_bl  = (const float*)d_in[pi++];
    const float* s10_Wr  = (const float*)d_in[pi++];   // (128,3)
    const float* s11_Wl  = (const float*)d_in[pi++];   // (128,128)
    const float* s11_bl  = (const float*)d_in[pi++];
    const float* s11_Wr  = (const float*)d_in[pi++];   // (128,128)
    const float* s20_Wl  = (const float*)d_in[pi++];   // (128,128)
    const float* s20_bl  = (const float*)d_in[pi++];
    const float* s20_Wr  = (const float*)d_in[pi++];   // (128,3)
    pi += 3;                                           // s2_1_* : dead (final x_f unused)
    const float* ln_g    = (const float*)d_in[pi++];
    const float* ln_b    = (const float*)d_in[pi++];
    const float* c1_W    = (const float*)d_in[pi++];   // (64,128,8) -> (64,1024)
    const float* c1_b    = (const float*)d_in[pi++];
    const float* c2_W    = (const float*)d_in[pi++];   // (64,64,8) -> (64,512)
    const float* c2_b    = (const float*)d_in[pi++];
    const float* c3_W    = (const float*)d_in[pi++];
    const float* c3_b    = (const float*)d_in[pi++];
    const float* d1_W    = (const float*)d_in[pi++];   // (100, 260800)
    const float* d1_b    = (const float*)d_in[pi++];
    const float* d2_W    = (const float*)d_in[pi++];   // (2,100)
    const float* d2_b    = (const float*)d_in[pi++];

    // ---- workspace carve ----
    char* ws = (char*)d_ws;
    size_t off = 0;
    auto carve = [&](size_t nflt) -> float* {
        off = (off + 255) & ~(size_t)255;
        float* p = (float*)(ws + off);
        off += nflt * sizeof(float);
        return p;
    };
    float* xp3     = carve((size_t)NPN * 3);      // initial x_p
    float* xf3     = carve((size_t)NFN * 3);      // initial x_f
    float* sumP3   = carve((size_t)NPN * 3);
    float* cntP    = carve((size_t)NPN);
    float* xpA     = carve((size_t)NPN * GD);     // x_p after s1_0 (then LN in place)
    float* sumF    = carve((size_t)NFN * GD);
    float* cntF    = carve((size_t)NFN);
    float* tmpF    = carve((size_t)NFN * GD);
    float* xfA     = carve((size_t)NFN * GD);     // x_f after s2_0 (then LN in place)
    float* sumP128 = carve((size_t)NPN * GD);
    float* xpB     = carve((size_t)NPN * GD);     // x_p after s1_1 -> conv input (16,128,4096)
    float* c1out   = carve((size_t)16 * 64 * LOUT1);
    float* c2out   = carve((size_t)16 * 64 * LOUT2);
    float* c3out   = carve((size_t)16 * 64 * LOUT3);
    float* d1acc   = carve((size_t)16 * 100);
    float* d1out   = carve((size_t)16 * 100);

    const dim3 blk(256);
    const int* row0 = edge;               // edge_index[0,:]
    const int* row1 = edge + 2097152;     // edge_index[1,:]

    // 1) x_p0 = (16,12288) @ pnode_W^T + b   -> (65536,3) flat
    gemm_nt_wmma<<<waveGrid(1LL * 768), blk, 0, stream>>>(
        x_dst, pnode_W, pnode_b, nullptr, nullptr, nullptr, xp3,
        16, 12288, 12288, 0, 0);

    // 2) x_f0 = (16,64) @ fnode_W^T + b      -> (1024,3) flat
    gemm_nt_wmma<<<waveGrid(12), blk, 0, stream>>>(
        x_src, fnode_W, fnode_b, nullptr, nullptr, nullptr, xf3,
        16, 192, 64, 0, 0);

    // 3) SAGE s1_0 (f->p, dims 3): scatter-mean then fused small combine
    zero_f32<<<elemGrid((long long)NPN * 3), blk, 0, stream>>>(sumP3, (long long)NPN * 3);
    zero_f32<<<elemGrid(NPN), blk, 0, stream>>>(cntP, NPN);
    scatter_mean3<<<elemGrid(E2N), blk, 0, stream>>>(xf3, row0, row1, 0, sumP3, cntP);
    sage_combine_small<<<elemGrid((long long)NPN * GD), blk, 0, stream>>>(
        sumP3, cntP, xp3, s10_Wl, s10_bl, s10_Wr, xpA, NPN);

    // 4) SAGE s2_0 (p->f): 128-dim scatter-mean + WMMA combine
    zero_f32<<<elemGrid((long long)NFN * GD), blk, 0, stream>>>(sumF, (long long)NFN * GD);
    zero_f32<<<elemGrid(NFN), blk, 0, stream>>>(cntF, NFN);
    scatter_mean128<<<elemGrid((long long)E2N * 8), blk, 0, stream>>>(
        xpA, row1, row0, 1, sumF, cntF);
    mean_div<<<elemGrid((long long)NFN * GD), blk, 0, stream>>>(sumF, cntF, NFN, GD);
    smallk_nt<<<elemGrid((long long)NFN * GD), blk, 0, stream>>>(xf3, s20_Wr, tmpF, NFN);
    gemm_nt_wmma<<<waveGrid(64LL * 8), blk, 0, stream>>>(
        sumF, s20_Wl, s20_bl, tmpF, nullptr, nullptr, xfA,
        NFN, GD, GD, 0, 1);

    // 5) LayerNorm (i==0 only), in place
    layernorm_rows<<<dim3(NPN), dim3(128), 0, stream>>>(xpA, ln_g, ln_b);
    layernorm_rows<<<dim3(NFN), dim3(128), 0, stream>>>(xfA, ln_g, ln_b);

    // 6) SAGE s1_1 (f->p, 128-dim): scatter-mean + fused dual-K WMMA combine
    zero_f32<<<elemGrid((long long)NPN * GD), blk, 0, stream>>>(sumP128, (long long)NPN * GD);
    zero_f32<<<elemGrid(NPN), blk, 0, stream>>>(cntP, NPN);
    scatter_mean128<<<elemGrid((long long)E2N * 8), blk, 0, stream>>>(
        xfA, row0, row1, 0, sumP128, cntP);
    mean_div<<<elemGrid((long long)NPN * GD), blk, 0, stream>>>(sumP128, cntP, NPN, GD);
    gemm_nt_wmma<<<waveGrid(4096LL * 8), blk, 0, stream>>>(
        sumP128, s11_Wl, s11_bl, nullptr, xpA, s11_Wr, xpB,
        NPN, GD, GD, GD, 1);
    // (SAGE s2_1 is dead: final x_f never feeds the output — skipped.)

    // 7) convs as implicit WMMA GEMMs; xpB flat == (16,128,4096)
    {
        long long w1 = (long long)((LOUT1 + 15) / 16) * 4;
        dim3 g1(waveGrid(w1).x, 16);
        conv1d_wmma<<<g1, blk, 0, stream>>>(xpB, c1_W, c1_b, c1out, 128, LIN0, LOUT1);
        long long w2 = (long long)((LOUT2 + 15) / 16) * 4;
        dim3 g2(waveGrid(w2).x, 16);
        conv1d_wmma<<<g2, blk, 0, stream>>>(c1out, c2_W, c2_b, c2out, 64, LOUT1, LOUT2);
        long long w3 = (long long)((LOUT3 + 15) / 16) * 4;
        dim3 g3(waveGrid(w3).x, 16);
        conv1d_wmma<<<g3, blk, 0, stream>>>(c2out, c3_W, c3_b, c3out, 64, LOUT2, LOUT3);
    }

    // 8) d1: (16,260800)@(100,260800)^T via split-K WMMA + atomic reduce
    zero_f32<<<elemGrid(1600), blk, 0, stream>>>(d1acc, 1600);
    {
        const int K = LOUT3 * 64;     // 260800
        const int CHUNK = 4096;
        const int chunks = (K + CHUNK - 1) / CHUNK;   // 64
        gemm_nt_splitk<<<waveGrid(7LL * chunks), blk, 0, stream>>>(
            c3out, d1_W, d1acc, 100, K, CHUNK);
    }
    bias_relu<<<elemGrid(1600), blk, 0, stream>>>(d1acc, d1_b, d1out, 16, 100);

    // 9) d2 + softmax -> (16,2)
    d2_softmax<<<dim3(1), dim3(32), 0, stream>>>(d1out, d2_W, d2_b, (float*)d_out);
}